// GraphInGraph_13151189860865
// MI455X (gfx1250) — compile-verified
//
#include <hip/hip_runtime.h>
#include <hip/hip_bf16.h>

// ---------------------------------------------------------------------------
// GraphInGraph pipeline for MI455X (gfx1250, wave32, WMMA).
// Dense stages use V_WMMA_F32_16X16X4_F32 fed from LDS (ds_load_b64) with
// bank-conflict padding; kNN Gram sweep keeps A fragments in VGPRs and
// prefetches the streaming operand (global_prefetch_b8).
// ---------------------------------------------------------------------------

typedef float v2f __attribute__((ext_vector_type(2)));
typedef float v8f __attribute__((ext_vector_type(8)));

// ---------------------------------------------------------------------------
// Utility: zero a float region
// ---------------------------------------------------------------------------
__global__ void k_zero(float* __restrict__ p, long long n) {
    long long t = (long long)blockIdx.x * blockDim.x + threadIdx.x;
    if (t < n) p[t] = 0.0f;
}

// ---------------------------------------------------------------------------
// Edge aggregation: agg[dst[e]][f] += x[src[e]][f]   (segment_sum over edges)
// ---------------------------------------------------------------------------
__global__ void k_edge_agg(const float* __restrict__ x, const int* __restrict__ src,
                           const int* __restrict__ dst, float* __restrict__ agg,
                           int E, int F) {
    long long t = (long long)blockIdx.x * blockDim.x + threadIdx.x;
    long long total = (long long)E * F;
    if (t >= total) return;
    int e = (int)(t / F);
    int f = (int)(t - (long long)e * F);
    atomicAdd(&agg[(long long)dst[e] * F + f], x[(long long)src[e] * F + f]);
}

// ---------------------------------------------------------------------------
// WMMA GEMM:  D = act( A0@B0 [+ A1@B1] + bias )
//   A row-major [M,K], B row-major [K,N], D row-major [M,N]; K%4==0, N<=64.
//   Block = 128 threads (4 waves): stages A tile(s) [16xK] and B^T tile(s)
//   [colsxK] into LDS with +2-float row padding (banks 64x4B: stride K+2
//   shifts each row's bank by 2 -> 16 distinct banks for the WMMA reads).
//   Each wave then computes a 16x16 tile via V_WMMA_F32_16X16X4_F32 fed by
//   aligned ds_load_b64 fragments.
//   act: 0 = none, 1 = relu, 2 = leaky_relu(0.01)
// A 16x4 fp32 layout (ISA 7.12.2): lanes 0-15 hold M=lane, VGPR0/1 = K{0,1};
// lanes 16-31 hold K{2,3}. B mirrored (lane = N, half selects K pair).
// C/D: lane&15 = N, VGPR r -> M = r + (lane>>4)*8.
// ---------------------------------------------------------------------------
__global__ void k_gemm_wmma(const float* __restrict__ A0, const float* __restrict__ B0,
                            const float* __restrict__ A1, const float* __restrict__ B1,
                            const float* __restrict__ bias, float* __restrict__ D,
                            int M, int K, int N, int act) {
    // worst cases: dual K=64 -> 2*16*66 + 2*64*66 = 10560 floats
    //              single K=128 -> 16*130 + 64*130 = 10400 floats
    __shared__ float smem[10688];
    const int  Kp   = K + 2;
    const bool dual = (A1 != nullptr);
    float* sA0 = smem;
    float* sA1 = sA0 + 16 * Kp;                 // valid only when dual
    float* sB0 = dual ? (sA1 + 16 * Kp) : sA1;
    float* sB1 = sB0 + 64 * Kp;                 // valid only when dual

    const int tid  = threadIdx.x;
    const int wave = tid >> 5;
    const int lane = tid & 31;
    const int half = lane >> 4;
    const int l    = lane & 15;
    const int m0    = blockIdx.y * 16;
    const int nbase = blockIdx.x * 64;
    int cols = N - nbase; if (cols > 64) cols = 64;

    // cooperative staging: A tiles (row-major, padded) ...
    for (int t = tid; t < 16 * K; t += 128) {
        int r = t / K, k = t - r * K;
        sA0[r * Kp + k] = A0[(long long)(m0 + r) * K + k];
        if (dual) sA1[r * Kp + k] = A1[(long long)(m0 + r) * K + k];
    }
    // ... and B transposed (B^T[n][k], padded) so fragments are contiguous
    for (int t = tid; t < cols * K; t += 128) {
        int nn = t / K, k = t - nn * K;
        sB0[nn * Kp + k] = B0[(long long)k * N + nbase + nn];
        if (dual) sB1[nn * Kp + k] = B1[(long long)k * N + nbase + nn];
    }
    __syncthreads();

    const int n0 = nbase + wave * 16;
    if (n0 >= N) return;      // wave-uniform; after the only barrier

    v8f acc = {};
    const int arow = l * Kp;
    const int brow = (wave * 16 + l) * Kp;
    for (int k0 = 0; k0 < K; k0 += 4) {
        int kk = k0 + half * 2;
        v2f a = *(const v2f*)&sA0[arow + kk];   // ds_load_b64, 8B aligned
        v2f b = *(const v2f*)&sB0[brow + kk];
        acc = __builtin_amdgcn_wmma_f32_16x16x4_f32(
            false, a, false, b, (short)0, acc, false, false);
    }
    if (dual) {
        for (int k0 = 0; k0 < K; k0 += 4) {
            int kk = k0 + half * 2;
            v2f a = *(const v2f*)&sA1[arow + kk];
            v2f b = *(const v2f*)&sB1[brow + kk];
            acc = __builtin_amdgcn_wmma_f32_16x16x4_f32(
                false, a, false, b, (short)0, acc, false, false);
        }
    }

    const int   n  = n0 + l;
    const float bv = (bias != nullptr) ? bias[n] : 0.0f;
    #pragma unroll
    for (int r = 0; r < 8; r++) {
        float v = acc[r] + bv;
        if (act == 1)      v = fmaxf(v, 0.0f);
        else if (act == 2) v = (v > 0.0f) ? v : 0.01f * v;
        D[(long long)(m0 + half * 8 + r) * N + n] = v;
    }
}

// ---------------------------------------------------------------------------
// Mean pool: gsum[batch[i]][f] += h[i][f]; gcnt[batch[i]] += 1
// ---------------------------------------------------------------------------
__global__ void k_pool_sum(const float* __restrict__ h, const int* __restrict__ batch,
                           float* __restrict__ gsum, float* __restrict__ gcnt,
                           int T, int F) {
    long long t = (long long)blockIdx.x * blockDim.x + threadIdx.x;
    long long total = (long long)T * F;
    if (t >= total) return;
    int i = (int)(t / F);
    int f = (int)(t - (long long)i * F);
    int b = batch[i];
    atomicAdd(&gsum[(long long)b * F + f], h[(long long)i * F + f]);
    if (f == 0) atomicAdd(&gcnt[b], 1.0f);
}

__global__ void k_pool_div(const float* __restrict__ gsum, const float* __restrict__ gcnt,
                           float* __restrict__ g, int NG, int F) {
    int t = blockIdx.x * blockDim.x + threadIdx.x;
    if (t >= NG * F) return;
    int b = t / F;
    g[t] = gsum[t] / fmaxf(gcnt[b], 1.0f);
}

// ---------------------------------------------------------------------------
// Row squared norms (d = 64)
// ---------------------------------------------------------------------------
__global__ void k_sqnorm(const float* __restrict__ g, float* __restrict__ sq,
                         int n, int d) {
    int i = blockIdx.x * blockDim.x + threadIdx.x;
    if (i >= n) return;
    float s = 0.0f;
    for (int f = 0; f < d; f++) { float v = g[(long long)i * d + f]; s += v * v; }
    sq[i] = s;
}

// ---------------------------------------------------------------------------
// Fused kNN (k=10) over 16384 x 64 features.
//  - block = 128 threads (4 waves), each block owns a 16-row tile.
//  - A fragments (loop-invariant 16-row block) preloaded into 16 v2f VGPR
//    pairs once; the column sweep only streams B rows (L2-resident, 4MB)
//    with a global_prefetch on the next tile.
//  - Each wave computes a 16x16 Gram sub-tile via WMMA, converts to
//    d2 = sq_m + sq_n - 2*dot in LDS; 8 threads/row keep register top-10
//    lists, merged through LDS at the end (includes self, like the ref).
// ---------------------------------------------------------------------------
__global__ void k_knn(const float* __restrict__ g, const float* __restrict__ sq,
                      int* __restrict__ knn, int n, int d /* must be 64 */) {
    __shared__ float lds_d2[16][64];
    __shared__ float md[16][80];
    __shared__ int   mi[16][80];

    const int wave = threadIdx.x >> 5;
    const int lane = threadIdx.x & 31;
    const int half = lane >> 4;
    const int l    = lane & 15;
    const int m0   = blockIdx.x * 16;

    const int row_t = threadIdx.x >> 3;   // 0..15 : row this thread scans
    const int seg   = threadIdx.x & 7;    // 0..7  : 8-column segment

    // preload this block's A fragments (16 k-steps of 16x4) into registers
    v2f af[16];
    {
        const float* Ap = &g[(long long)(m0 + l) * 64 + half * 2];
        #pragma unroll
        for (int k = 0; k < 16; k++) af[k] = *(const v2f*)&Ap[k * 4];
    }

    float bd[10]; int bi[10];
    #pragma unroll
    for (int q = 0; q < 10; q++) { bd[q] = 3.4e38f; bi[q] = 0; }
    float worst = 3.4e38f; int wslot = 0;

    for (int j0 = 0; j0 < n; j0 += 64) {
        int n0 = j0 + wave * 16;
        if (j0 + 64 < n)  // uniform: hint next streaming tile into cache
            __builtin_prefetch(&g[(long long)(j0 + 64 + wave * 16 + l) * 64], 0, 0);

        v8f acc = {};
        const float* Bp = &g[(long long)(n0 + l) * 64 + half * 2];
        #pragma unroll
        for (int k = 0; k < 16; k++) {
            v2f b = *(const v2f*)&Bp[k * 4];
            acc = __builtin_amdgcn_wmma_f32_16x16x4_f32(
                false, af[k], false, b, (short)0, acc, false, false);
        }
        #pragma unroll
        for (int r = 0; r < 8; r++) {
            int row = half * 8 + r;
            lds_d2[row][wave * 16 + l] = sq[m0 + row] + sq[n0 + l] - 2.0f * acc[r];
        }
        __syncthreads();

        // scan 8 columns of this 64-wide tile for my row
        #pragma unroll
        for (int c8 = 0; c8 < 8; c8++) {
            int c = seg * 8 + c8;
            float dv = lds_d2[row_t][c];
            if (dv < worst) {
                bd[wslot] = dv; bi[wslot] = j0 + c;
                worst = bd[0]; wslot = 0;
                #pragma unroll
                for (int q = 1; q < 10; q++)
                    if (bd[q] > worst) { worst = bd[q]; wslot = q; }
            }
        }
        __syncthreads();
    }

    // merge 8 per-thread top-10 lists per row
    #pragma unroll
    for (int q = 0; q < 10; q++) {
        md[row_t][seg * 10 + q] = bd[q];
        mi[row_t][seg * 10 + q] = bi[q];
    }
    __syncthreads();
    if (threadIdx.x < 16) {
        int row = threadIdx.x;
        for (int q = 0; q < 10; q++) {
            float best = 3.4e38f; int bj = 0;
            for (int t = 0; t < 80; t++)
                if (md[row][t] < best) { best = md[row][t]; bj = t; }
            knn[(long long)(m0 + row) * 10 + q] = mi[row][bj];
            md[row][bj] = 3.4e38f;
        }
    }
}

// ---------------------------------------------------------------------------
// Build edge-conv message inputs: xcat[(i*k+q)] = [ g[i], g[nbr]-g[i] ]
// ---------------------------------------------------------------------------
__global__ void k_build_xcat(const float* __restrict__ g, const int* __restrict__ knn,
                             float* __restrict__ xcat, int n, int d, int k) {
    long long t = (long long)blockIdx.x * blockDim.x + threadIdx.x;
    long long total = (long long)n * k * d;
    if (t >= total) return;
    int f  = (int)(t % d);
    long long rq = t / d;
    int q  = (int)(rq % k);
    int i  = (int)(rq / k);
    float xi = g[(long long)i * d + f];
    float xj = g[(long long)knn[(long long)i * k + q] * d + f];
    long long row = (long long)i * k + q;
    xcat[row * 2 * d + f]     = xi;
    xcat[row * 2 * d + d + f] = xj - xi;
}

// ---------------------------------------------------------------------------
// Max over k neighbors, then outer ReLU:  out[i][f] = relu(max_q msg[i*k+q][f])
// ---------------------------------------------------------------------------
__global__ void k_maxpool_relu(const float* __restrict__ msg, float* __restrict__ out,
                               int n, int d, int k) {
    int t = blockIdx.x * blockDim.x + threadIdx.x;
    if (t >= n * d) return;
    int f = t % d;
    int i = t / d;
    float m = -3.4e38f;
    for (int q = 0; q < k; q++)
        m = fmaxf(m, msg[((long long)i * k + q) * d + f]);
    out[t] = fmaxf(m, 0.0f);
}

// ---------------------------------------------------------------------------
// Final narrow GEMM (N=10): out = a @ W + b
// ---------------------------------------------------------------------------
__global__ void k_fc2(const float* __restrict__ a, const float* __restrict__ W,
                      const float* __restrict__ b, float* __restrict__ out,
                      int n, int K, int O) {
    int t = blockIdx.x * blockDim.x + threadIdx.x;
    if (t >= n * O) return;
    int o = t % O;
    int i = t / O;
    float s = b[o];
    for (int f = 0; f < K; f++)
        s += a[(long long)i * K + f] * W[(long long)f * O + o];
    out[t] = s;
}

// ---------------------------------------------------------------------------
// Host launch
// ---------------------------------------------------------------------------
extern "C" void kernel_launch(void* const* d_in, const int* in_sizes, int n_in,
                              void* d_out, int out_size, void* d_ws, size_t ws_size,
                              hipStream_t stream) {
    (void)n_in; (void)out_size; (void)ws_size;

    const float* x     = (const float*)d_in[0];
    const int*   ei    = (const int*)  d_in[1];
    const int*   batch = (const int*)  d_in[2];
    const float* W1r   = (const float*)d_in[3];
    const float* W1e   = (const float*)d_in[4];
    const float* b1    = (const float*)d_in[5];
    const float* W2r   = (const float*)d_in[6];
    const float* W2e   = (const float*)d_in[7];
    const float* b2    = (const float*)d_in[8];
    const float* Wd1   = (const float*)d_in[9];
    const float* bd1   = (const float*)d_in[10];
    const float* Wd2   = (const float*)d_in[11];
    const float* bd2   = (const float*)d_in[12];
    const float* Wf1   = (const float*)d_in[13];
    const float* bf1   = (const float*)d_in[14];
    const float* Wf2   = (const float*)d_in[15];
    const float* bf2   = (const float*)d_in[16];

    const int T  = in_sizes[0] / 32;   // 200000
    const int E  = in_sizes[1] / 2;    // 1600000
    const int NG = 16384, H = 64, KNN = 10, FC = 32, OUT = 10;
    const int* src = ei;
    const int* dst = ei + E;

    // bump allocator over d_ws (256B aligned chunks)
    char*  wsb = (char*)d_ws;
    size_t off = 0;
    auto alloc = [&](size_t bytes) -> void* {
        void* p = wsb + off;
        off = (off + bytes + 255) & ~(size_t)255;
        return p;
    };
    float* agg  = (float*)alloc((size_t)T * H * 4);
    float* h1   = (float*)alloc((size_t)T * H * 4);
    float* h2   = (float*)alloc((size_t)T * H * 4);
    float* gsum = (float*)alloc((size_t)NG * H * 4);
    float* gcnt = (float*)alloc((size_t)NG * 4);
    float* g0   = (float*)alloc((size_t)NG * H * 4);
    float* sq   = (float*)alloc((size_t)NG * 4);
    int*   idx  = (int*)  alloc((size_t)NG * KNN * 4);
    float* xcat = (float*)alloc((size_t)NG * KNN * 2 * H * 4);
    float* msg  = (float*)alloc((size_t)NG * KNN * H * 4);
    float* gA   = (float*)alloc((size_t)NG * H * 4);
    float* gB   = (float*)alloc((size_t)NG * H * 4);
    float* fc1  = (float*)alloc((size_t)NG * FC * 4);

    auto cdiv = [](long long a, long long b) { return (int)((a + b - 1) / b); };

    // ---- GraphConv layer 1: h1 = relu(x@W1r + agg@W1e + b1) ----
    k_zero<<<cdiv((long long)T * H, 256), 256, 0, stream>>>(agg, (long long)T * H);
    k_edge_agg<<<cdiv((long long)E * 32, 256), 256, 0, stream>>>(x, src, dst, agg, E, 32);
    k_gemm_wmma<<<dim3(1, T / 16), 128, 0, stream>>>(x, W1r, agg, W1e, b1, h1, T, 32, H, 1);

    // ---- GraphConv layer 2: h2 = relu(h1@W2r + agg@W2e + b2) ----
    k_zero<<<cdiv((long long)T * H, 256), 256, 0, stream>>>(agg, (long long)T * H);
    k_edge_agg<<<cdiv((long long)E * 64, 256), 256, 0, stream>>>(h1, src, dst, agg, E, 64);
    k_gemm_wmma<<<dim3(1, T / 16), 128, 0, stream>>>(h1, W2r, agg, W2e, b2, h2, T, 64, H, 1);

    // ---- mean pool -> g0 [NG,64] ----
    k_zero<<<cdiv((long long)NG * H, 256), 256, 0, stream>>>(gsum, (long long)NG * H);
    k_zero<<<cdiv((long long)NG, 256), 256, 0, stream>>>(gcnt, (long long)NG);
    k_pool_sum<<<cdiv((long long)T * H, 256), 256, 0, stream>>>(h2, batch, gsum, gcnt, T, H);
    k_pool_div<<<cdiv((long long)NG * H, 256), 256, 0, stream>>>(gsum, gcnt, g0, NG, H);

    // ---- DynamicEdgeConv 1 ----
    k_sqnorm<<<cdiv(NG, 256), 256, 0, stream>>>(g0, sq, NG, H);
    k_knn<<<NG / 16, 128, 0, stream>>>(g0, sq, idx, NG, H);
    k_build_xcat<<<cdiv((long long)NG * KNN * H, 256), 256, 0, stream>>>(g0, idx, xcat, NG, H, KNN);
    k_gemm_wmma<<<dim3(1, (NG * KNN) / 16), 128, 0, stream>>>(
        xcat, Wd1, nullptr, nullptr, bd1, msg, NG * KNN, 2 * H, H, 2);
    k_maxpool_relu<<<cdiv((long long)NG * H, 256), 256, 0, stream>>>(msg, gA, NG, H, KNN);

    // ---- DynamicEdgeConv 2 ----
    k_sqnorm<<<cdiv(NG, 256), 256, 0, stream>>>(gA, sq, NG, H);
    k_knn<<<NG / 16, 128, 0, stream>>>(gA, sq, idx, NG, H);
    k_build_xcat<<<cdiv((long long)NG * KNN * H, 256), 256, 0, stream>>>(gA, idx, xcat, NG, H, KNN);
    k_gemm_wmma<<<dim3(1, (NG * KNN) / 16), 128, 0, stream>>>(
        xcat, Wd2, nullptr, nullptr, bd2, msg, NG * KNN, 2 * H, H, 2);
    k_maxpool_relu<<<cdiv((long long)NG * H, 256), 256, 0, stream>>>(msg, gB, NG, H, KNN);

    // ---- classifier: fc1 = relu(gB@Wf1 + bf1);  out = fc1@Wf2 + bf2 ----
    k_gemm_wmma<<<dim3(1, NG / 16), 128, 0, stream>>>(
        gB, Wf1, nullptr, nullptr, bf1, fc1, NG, H, FC, 1);
    k_fc2<<<cdiv((long long)NG * OUT, 256), 256, 0, stream>>>(
        fc1, Wf2, bf2, (float*)d_out, NG, FC, OUT);
}